// HeteroRelationalGraphConv_26577257628122
// MI455X (gfx1250) — compile-verified
//
#include <hip/hip_runtime.h>

// ---------------------------------------------------------------------------
// HeteroRelationalGraphConv on MI455X (gfx1250, wave32).
//   m_i = X_i @ W_i + b_i   (fp32 WMMA 16x16x4, LDS-staged, conflict-free)
//   h   = relu(segment_sum(...)) chain via L2 atomics
// Dual-weight GEMM fuses the two x_user matmuls (one X staging, A-fragment
// reused for two WMMAs). Scatter prefetches next random gather row
// (global_prefetch_b8) while issuing the current row's f32 atomics.
// ---------------------------------------------------------------------------

#define D_FEAT 128

typedef __attribute__((ext_vector_type(2))) float v2f;
typedef __attribute__((ext_vector_type(8))) float v8f;

#define XS_STRIDE 132   // 132 % 64 banks -> 4-bank row skew: A-frag conflict-free
#define WS_STRIDE 136   // 136 % 64 banks -> 8-bank row skew: B-frag conflict-free

// ---------------------------------------------------------------------------
// Dual GEMM: out0 = X@W0 + b0, out1 = X@W1 + b1. Block = 256 thr (8 waves),
// block tile = 128 rows; wave w owns rows [16w,16w+16), loops 8 col-tiles.
// ---------------------------------------------------------------------------
__global__ __launch_bounds__(256) void gemm128_bias_wmma_dual(
    const float* __restrict__ X,
    const float* __restrict__ W0, const float* __restrict__ b0, float* __restrict__ out0,
    const float* __restrict__ W1, const float* __restrict__ b1, float* __restrict__ out1,
    int nrows) {
  __shared__ float Xs[128 * XS_STRIDE];
  __shared__ float Ws0[128 * WS_STRIDE];
  __shared__ float Ws1[128 * WS_STRIDE];

  const int tid  = threadIdx.x;
  const int wave = tid >> 5;
  const int lane = tid & 31;
  const int half = lane >> 4;
  const int l16  = lane & 15;
  const int mbase = blockIdx.x * 128;
  const bool fullTile = (mbase + 128 <= nrows);

  for (int i = tid; i < 128 * 32; i += 256) {
    const int r = i >> 5;
    const int c = (i & 31) << 2;
    const int gr = mbase + r;
    float4 vx = make_float4(0.f, 0.f, 0.f, 0.f);
    if (gr < nrows) vx = *(const float4*)(X + (size_t)gr * D_FEAT + c);
    *(float4*)(&Xs[r * XS_STRIDE + c]) = vx;
    *(float4*)(&Ws0[r * WS_STRIDE + c]) = *(const float4*)(W0 + (size_t)r * D_FEAT + c);
    *(float4*)(&Ws1[r * WS_STRIDE + c]) = *(const float4*)(W1 + (size_t)r * D_FEAT + c);
  }
  __syncthreads();

  // V_WMMA_F32_16X16X4_F32 fragments (wave32):
  //  A 16x4 : lane l16 = row, VGPR0/1 = K {0,1} (lanes 0-15) / {2,3} (16-31)
  //  B 4x16 : lane l16 = col, VGPR r lanes 0-15 = K row r, lanes 16-31 = r+2
  //  C 16x16: VGPR j lanes 0-15 = M j, lanes 16-31 = M j+8
  const int arow = (wave << 4) + l16;

  for (int nt = 0; nt < 8; ++nt) {
    const int col = (nt << 4) + l16;
    v8f acc0 = {};
    v8f acc1 = {};
#pragma unroll 8
    for (int k = 0; k < 128; k += 4) {
      const int ka = k + (half << 1);
      const v2f a = *(const v2f*)(&Xs[arow * XS_STRIDE + ka]);  // ds_load_b64
      v2f bA, bB;
      bA.x = Ws0[ka * WS_STRIDE + col];
      bA.y = Ws0[(ka + 1) * WS_STRIDE + col];
      bB.x = Ws1[ka * WS_STRIDE + col];
      bB.y = Ws1[(ka + 1) * WS_STRIDE + col];
      acc0 = __builtin_amdgcn_wmma_f32_16x16x4_f32(false, a, false, bA,
                                                   (short)0, acc0, false, false);
      acc1 = __builtin_amdgcn_wmma_f32_16x16x4_f32(false, a, false, bB,
                                                   (short)0, acc1, false, false);
    }
    const float bv0 = b0[col];
    const float bv1 = b1[col];
    const int rbase = mbase + (wave << 4) + (half << 3);
    if (fullTile) {
#pragma unroll
      for (int j = 0; j < 8; ++j) {
        out0[(size_t)(rbase + j) * D_FEAT + col] = acc0[j] + bv0;
        out1[(size_t)(rbase + j) * D_FEAT + col] = acc1[j] + bv1;
      }
    } else {
#pragma unroll
      for (int j = 0; j < 8; ++j) {
        if (rbase + j < nrows) {
          out0[(size_t)(rbase + j) * D_FEAT + col] = acc0[j] + bv0;
          out1[(size_t)(rbase + j) * D_FEAT + col] = acc1[j] + bv1;
        }
      }
    }
  }
}

// ---------------------------------------------------------------------------
// Single GEMM (item relation): out = X@W + b.
// ---------------------------------------------------------------------------
__global__ __launch_bounds__(256) void gemm128_bias_wmma(
    const float* __restrict__ X, const float* __restrict__ W,
    const float* __restrict__ bias, float* __restrict__ out, int nrows) {
  __shared__ float Xs[128 * XS_STRIDE];
  __shared__ float Ws[128 * WS_STRIDE];

  const int tid  = threadIdx.x;
  const int wave = tid >> 5;
  const int lane = tid & 31;
  const int half = lane >> 4;
  const int l16  = lane & 15;
  const int mbase = blockIdx.x * 128;
  const bool fullTile = (mbase + 128 <= nrows);

  for (int i = tid; i < 128 * 32; i += 256) {
    const int r = i >> 5;
    const int c = (i & 31) << 2;
    const int gr = mbase + r;
    float4 vx = make_float4(0.f, 0.f, 0.f, 0.f);
    if (gr < nrows) vx = *(const float4*)(X + (size_t)gr * D_FEAT + c);
    *(float4*)(&Xs[r * XS_STRIDE + c]) = vx;
    *(float4*)(&Ws[r * WS_STRIDE + c]) = *(const float4*)(W + (size_t)r * D_FEAT + c);
  }
  __syncthreads();

  const int arow = (wave << 4) + l16;

  for (int nt = 0; nt < 8; ++nt) {
    const int col = (nt << 4) + l16;
    v8f acc = {};
#pragma unroll 8
    for (int k = 0; k < 128; k += 4) {
      const int ka = k + (half << 1);
      const v2f a = *(const v2f*)(&Xs[arow * XS_STRIDE + ka]);
      v2f b;
      b.x = Ws[ka * WS_STRIDE + col];
      b.y = Ws[(ka + 1) * WS_STRIDE + col];
      acc = __builtin_amdgcn_wmma_f32_16x16x4_f32(false, a, false, b,
                                                  (short)0, acc, false, false);
    }
    const float bv = bias[col];
    const int rbase = mbase + (wave << 4) + (half << 3);
    if (fullTile) {
#pragma unroll
      for (int j = 0; j < 8; ++j)
        out[(size_t)(rbase + j) * D_FEAT + col] = acc[j] + bv;
    } else {
#pragma unroll
      for (int j = 0; j < 8; ++j)
        if (rbase + j < nrows) out[(size_t)(rbase + j) * D_FEAT + col] = acc[j] + bv;
    }
  }
}

// ---------------------------------------------------------------------------
// Edge scatter: acc[dst[e], :] += msg[src[e], :]. One wave per edge, lane =
// 4 contiguous floats (coalesced 512B row read, 4 f32 atomics per lane at L2).
// Prefetch next iteration's random gather row while atomics drain.
// ---------------------------------------------------------------------------
__global__ __launch_bounds__(256) void scatter_add_rows(
    const float* __restrict__ msg, const int* __restrict__ src,
    const int* __restrict__ dst, float* __restrict__ acc, int nE) {
  const int lane   = threadIdx.x & 31;
  const int waveId = (blockIdx.x * 256 + threadIdx.x) >> 5;
  const int nWaves = (gridDim.x * 256) >> 5;
  const int c4 = lane << 2;
  for (int e = waveId; e < nE; e += nWaves) {
    const int s = __builtin_amdgcn_readfirstlane(src[e]);  // scalarize indices
    const int d = __builtin_amdgcn_readfirstlane(dst[e]);
    const int eN = e + nWaves;
    if (eN < nE) {  // prefetch next gather row (random access -> L2 latency)
      const int sN = __builtin_amdgcn_readfirstlane(src[eN]);
      __builtin_prefetch(msg + (size_t)sN * D_FEAT + c4, 0, 1);
    }
    const float4 m = *(const float4*)(msg + (size_t)s * D_FEAT + c4);
    float* p = acc + (size_t)d * D_FEAT + c4;
    atomicAdd(p + 0, m.x);
    atomicAdd(p + 1, m.y);
    atomicAdd(p + 2, m.z);
    atomicAdd(p + 3, m.w);
  }
}

// ---------------------------------------------------------------------------
// Elementwise helpers (float4-vectorized).
// ---------------------------------------------------------------------------
__global__ __launch_bounds__(256) void zero4(float4* __restrict__ p, int n4) {
  const int i = blockIdx.x * 256 + threadIdx.x;
  if (i < n4) p[i] = make_float4(0.f, 0.f, 0.f, 0.f);
}

__global__ __launch_bounds__(256) void relu4(float4* __restrict__ p, int n4) {
  const int i = blockIdx.x * 256 + threadIdx.x;
  if (i < n4) {
    float4 v = p[i];
    v.x = fmaxf(v.x, 0.f);
    v.y = fmaxf(v.y, 0.f);
    v.z = fmaxf(v.z, 0.f);
    v.w = fmaxf(v.w, 0.f);
    p[i] = v;
  }
}

// ---------------------------------------------------------------------------
extern "C" void kernel_launch(void* const* d_in, const int* in_sizes, int n_in,
                              void* d_out, int out_size, void* d_ws, size_t ws_size,
                              hipStream_t stream) {
  const float* x_user = (const float*)d_in[0];
  const float* x_item = (const float*)d_in[1];
  const float* W0 = (const float*)d_in[2];
  const float* b0 = (const float*)d_in[3];
  const float* W1 = (const float*)d_in[4];
  const float* b1 = (const float*)d_in[5];
  const float* W2 = (const float*)d_in[6];
  const float* b2 = (const float*)d_in[7];
  const int* src0 = (const int*)d_in[8];
  const int* dst0 = (const int*)d_in[9];
  const int* src1 = (const int*)d_in[10];
  const int* dst1 = (const int*)d_in[11];
  const int* src2 = (const int*)d_in[12];
  const int* dst2 = (const int*)d_in[13];

  const int nUser = in_sizes[0] / D_FEAT;   // 100000
  const int nItem = in_sizes[1] / D_FEAT;   // 50000
  const int nE    = in_sizes[8];            // 500000

  float* out = (float*)d_out;               // [(nUser+nItem) x 128]
  float* m0 = (float*)d_ws;                 // [nUser x 128]
  float* m1 = m0 + (size_t)nUser * D_FEAT;  // [nUser x 128]
  float* m2 = m1 + (size_t)nUser * D_FEAT;  // [nItem x 128]
  float* out_item = out + (size_t)nUser * D_FEAT;

  const int n4_all  = (nUser + nItem) * (D_FEAT / 4);
  const int n4_user = nUser * (D_FEAT / 4);

  // 1) zero accumulators (d_out is poisoned, so must clear every call)
  zero4<<<(n4_all + 255) / 256, 256, 0, stream>>>((float4*)out, n4_all);

  // 2) dense messages via fp32 WMMA (x_user staged once for W0 and W1)
  const int gU = (nUser + 127) / 128;
  const int gI = (nItem + 127) / 128;
  gemm128_bias_wmma_dual<<<gU, 256, 0, stream>>>(x_user, W0, b0, m0,
                                                 W1, b1, m1, nUser);
  gemm128_bias_wmma<<<gI, 256, 0, stream>>>(x_item, W2, b2, m2, nItem);

  const int scatterBlocks = 2048;  // 16K waves grid-striding 500K edges

  // 3) rel0: user->user into user accumulator, then relu (reference order)
  scatter_add_rows<<<scatterBlocks, 256, 0, stream>>>(m0, src0, dst0, out, nE);
  relu4<<<(n4_user + 255) / 256, 256, 0, stream>>>((float4*)out, n4_user);

  // 4) rel2: item->user on top of relu'd h_user; rel1: user->item
  scatter_add_rows<<<scatterBlocks, 256, 0, stream>>>(m2, src2, dst2, out, nE);
  scatter_add_rows<<<scatterBlocks, 256, 0, stream>>>(m1, src1, dst1, out_item, nE);

  // 5) final relu over both node types
  relu4<<<(n4_all + 255) / 256, 256, 0, stream>>>((float4*)out, n4_all);
}